// Aggregator_2714419331492
// MI455X (gfx1250) — compile-verified
//
#include <hip/hip_runtime.h>

#define H_DIM   128
#define H2      256                    // 2*H_DIM
#define NEIGH   200
#define BATCH   512
#define SEQL    10
#define GROUPS  (BATCH * SEQL)         // 5120
#define OUTW    (3 * H_DIM)            // 384

typedef __attribute__((ext_vector_type(2))) float v2f;
typedef __attribute__((ext_vector_type(4))) float v4f;
typedef __attribute__((ext_vector_type(8))) float v8f;

// ---------------------------------------------------------------------------
// Kernel 1: weighted segment mean.  mean[g, :] = (1/200) * sum_i w[g*200+i] *
// embeds[g*200+i, :].   One block per group; 256 threads; thread t handles
// column quad (t%64)*4..+3 over row slice t/64, t/64+4, ... (50 rows each).
// embeds_stack is a 1.05 GB read-once stream (>> 192 MB L2), so use
// non-temporal loads to keep mean/W resident in L2 for the WMMA stage.
// ---------------------------------------------------------------------------
__global__ __launch_bounds__(256) void seg_mean_kernel(
    const float* __restrict__ embeds,   // [GROUPS*NEIGH, 256]
    const float* __restrict__ relw,     // [GROUPS*NEIGH]
    float* __restrict__ mean)           // [GROUPS, 256]  (workspace)
{
    __shared__ float sw[NEIGH];
    __shared__ v4f   part[256];

    const int g = blockIdx.x;
    const int t = threadIdx.x;

    if (t < NEIGH)
        sw[t] = __builtin_nontemporal_load(relw + (size_t)g * NEIGH + t);
    __syncthreads();

    const int q = t & 63;     // column quad index (0..63)
    const int r = t >> 6;     // row slice (0..3)

    const float* base = embeds + (size_t)g * NEIGH * H2 + q * 4;

    v4f acc = {0.f, 0.f, 0.f, 0.f};
    #pragma unroll 5
    for (int i = r; i < NEIGH; i += 4) {
        const float wv = sw[i];
        const v4f e = __builtin_nontemporal_load((const v4f*)(base + (size_t)i * H2));
        acc += e * wv;
    }
    part[t] = acc;
    __syncthreads();

    if (t < 64) {
        v4f s = part[t] + part[t + 64] + part[t + 128] + part[t + 192];
        s *= (1.0f / (float)NEIGH);
        *(v4f*)(mean + (size_t)g * H2 + t * 4) = s;
    }
}

// ---------------------------------------------------------------------------
// Kernel 2: h = relu(mean @ W^T + b) via V_WMMA_F32_16X16X4_F32, with the
// result written straight into d_out[:, 0:256] at row stride 384.
//
// One wave computes one 16x16 output tile, K=256 -> 64 WMMA ops per wave.
// A-frag layout (32-bit A 16x4): lane L holds row m0 + L%16, K pair 2*(L/16).
// B-frag (4x16): lane L holds col n0 + L%16, K pair 2*(L/16); since
// B[k][n] = W[n][k], each lane loads two consecutive floats of W row n.
// C/D: lane L, vgpr r  <->  (m0 + r + 8*(L/16), n0 + L%16).
// ---------------------------------------------------------------------------
__global__ __launch_bounds__(256) void gemm_relu_kernel(
    const float* __restrict__ A,     // mean [GROUPS, 256]
    const float* __restrict__ W,     // [256, 256] row-major
    const float* __restrict__ bias,  // [256]
    float* __restrict__ out)         // [GROUPS, 384]; writes cols 0..255
{
    const int lane = threadIdx.x & 31;
    const int wave = threadIdx.x >> 5;
    const int tile = blockIdx.x * 8 + wave;       // 5120 tiles total
    const int m0   = (tile >> 4) << 4;            // 320 tiles along M
    const int n0   = (tile & 15) << 4;            // 16 tiles along N
    const int lg   = lane >> 4;                   // 0 or 1 (K-pair group)
    const int lm   = lane & 15;

    const float* arow = A + (size_t)(m0 + lm) * H2 + 2 * lg;
    const float* brow = W + (size_t)(n0 + lm) * H2 + 2 * lg;

    v8f acc = {};
    #pragma unroll 8
    for (int k = 0; k < H2; k += 4) {
        v2f a = *(const v2f*)(arow + k);
        v2f b = *(const v2f*)(brow + k);
        acc = __builtin_amdgcn_wmma_f32_16x16x4_f32(
            /*neg_a=*/false, a, /*neg_b=*/false, b,
            /*c_mod=*/(short)0, acc, /*reuse_a=*/false, /*reuse_b=*/false);
    }

    const int   n  = n0 + lm;
    const float bv = bias[n];
    #pragma unroll
    for (int r = 0; r < 8; ++r) {
        const int m = m0 + r + 8 * lg;
        float v = acc[r] + bv;
        v = v > 0.f ? v : 0.f;
        out[(size_t)m * OUTW + n] = v;
    }
}

// ---------------------------------------------------------------------------
// Kernel 3: out[b, s, 256:384] = ent_embeds[s_tem[b], :] for all s.
// One block (128 threads) per batch element: each thread gathers one float of
// the entity row once, then broadcasts it to all 10 timesteps from registers.
// ---------------------------------------------------------------------------
__global__ __launch_bounds__(128) void ent_fill_kernel(
    const float* __restrict__ ent,    // [NUM_ENTS, 128]
    const int*   __restrict__ s_tem,  // [BATCH]
    float* __restrict__ out)          // [BATCH, SEQL, 384]
{
    const int b = blockIdx.x;
    const int t = threadIdx.x;        // 0..127
    const float v = ent[(size_t)s_tem[b] * H_DIM + t];
    float* dst = out + (size_t)b * SEQL * OUTW + 2 * H_DIM + t;

    #pragma unroll
    for (int s = 0; s < SEQL; ++s)
        dst[(size_t)s * OUTW] = v;
}

// ---------------------------------------------------------------------------
// Host-side launch.  Input order (from setup_inputs):
//  0 embeds_stack [TOTAL,256] f32   1 rel_weights [TOTAL] f32
//  2 ent_embeds [NUM_ENTS,128] f32  3 W [256,256] f32   4 b [256] f32
//  5 length [G] i32  6 segment_ids [TOTAL] i32  7 len_non_zero [B] i32
//  8 s_tem [B] i32
// Output: [512, 10, 384] f32.
// Workspace: mean [5120, 256] f32 = 5.24 MB.
// ---------------------------------------------------------------------------
extern "C" void kernel_launch(void* const* d_in, const int* in_sizes, int n_in,
                              void* d_out, int out_size, void* d_ws, size_t ws_size,
                              hipStream_t stream) {
    (void)in_sizes; (void)n_in; (void)out_size; (void)ws_size;

    const float* embeds = (const float*)d_in[0];
    const float* relw   = (const float*)d_in[1];
    const float* ent    = (const float*)d_in[2];
    const float* W      = (const float*)d_in[3];
    const float* bias   = (const float*)d_in[4];
    const int*   s_tem  = (const int*)d_in[8];

    float* out  = (float*)d_out;
    float* mean = (float*)d_ws;           // [GROUPS, 256]

    seg_mean_kernel<<<GROUPS, 256, 0, stream>>>(embeds, relw, mean);
    gemm_relu_kernel<<<GROUPS / 8, 256, 0, stream>>>(mean, W, bias, out);
    ent_fill_kernel<<<BATCH, 128, 0, stream>>>(ent, s_tem, out);
}